// NERModel_15272903705062
// MI455X (gfx1250) — compile-verified
//
#include <hip/hip_runtime.h>
#include <hip/hip_bf16.h>

typedef __attribute__((ext_vector_type(2))) float v2f;
typedef __attribute__((ext_vector_type(8))) float v8f;

#define S_LEN    512
#define BATCH    64
#define HIDDEN   768
#define TAGS     32      // 30 labels + start + end
#define START_TAG 30
#define END_TAG   31
#define NEG_LOGIT (-100000.0f)

// ---------------------------------------------------------------------------
// Kernel A: emissions GEMM  emit[s,b,t] = emb[s,b,:] @ W[:,t] + b[t]
// Rows = S*B = 32768, K = 768, N = 32.  Memory-bound (~105 MB @ 23.3 TB/s),
// so we use the exact-precision f32 WMMA path: V_WMMA_F32_16X16X4_F32.
// One wave computes a 16x32 output tile: 192 K-steps x 2 N-tiles = 384 WMMAs.
// ---------------------------------------------------------------------------
__global__ __launch_bounds__(256) void emit_gemm_kernel(
    const float* __restrict__ emb,   // [S*B, H] row-major
    const float* __restrict__ W,     // [H, T]  row-major
    const float* __restrict__ bias,  // [T]
    float* __restrict__ emit)        // [S*B, T] row-major
{
  const int lane = threadIdx.x & 31;
  const int wave = threadIdx.x >> 5;
  const int row0 = (blockIdx.x * 8 + wave) * 16;   // 256 blocks * 8 waves * 16 rows = 32768
  const int half = lane >> 4;                      // 0: K pair {k,k+1}, 1: {k+2,k+3}
  const int l15  = lane & 15;

  // A fragment (16x4 f32): lane l<16 holds A[l][k..k+1]; lane l+16 holds A[l][k+2..k+3]
  const float* aBase = emb + (size_t)(row0 + l15) * HIDDEN + 2 * half;
  // B fragment (4x16 f32): lane n<16 holds W[k][n],W[k+1][n]; lane n+16 holds W[k+2][n],W[k+3][n]
  const float* wBase = W + (size_t)(2 * half) * TAGS + l15;

  v8f acc0 = {};   // N-tile 0 (cols 0..15)
  v8f acc1 = {};   // N-tile 1 (cols 16..31)

#pragma unroll 8
  for (int k = 0; k < HIDDEN; k += 4) {
    v2f a = *(const v2f*)(aBase + k);              // global_load_b64
    const float* wk = wBase + (size_t)k * TAGS;
    v2f b0, b1;
    b0.x = wk[0];          b0.y = wk[TAGS];
    b1.x = wk[16];         b1.y = wk[TAGS + 16];
    acc0 = __builtin_amdgcn_wmma_f32_16x16x4_f32(false, a, false, b0, (short)0, acc0, false, false);
    acc1 = __builtin_amdgcn_wmma_f32_16x16x4_f32(false, a, false, b1, (short)0, acc1, false, false);
  }

  // D layout: lanes 0-15 -> cols 0..15 rows M=0..7 in v0..7; lanes 16-31 -> rows M=8..15
  const float bias0 = bias[l15];
  const float bias1 = bias[l15 + 16];
  float* outRow = emit + (size_t)(row0 + half * 8) * TAGS;
#pragma unroll
  for (int i = 0; i < 8; ++i) {
    outRow[(size_t)i * TAGS + l15]      = acc0[i] + bias0;
    outRow[(size_t)i * TAGS + l15 + 16] = acc1[i] + bias1;
  }
}

// ---------------------------------------------------------------------------
// Kernel B: CRF forward (log-partition), gold score, Viterbi + backtrace.
// One block per batch element; 2 waves: wave0 = alpha recursion + gold score,
// wave1 = Viterbi recursion + backtrace.  lane == tag (T == wave32 width).
// dp broadcast via v_readlane (__shfl with constant lane), backpointers in LDS.
// No barriers needed: the two waves touch disjoint state and outputs.
// ---------------------------------------------------------------------------
__global__ __launch_bounds__(64) void crf_kernel(
    const float* __restrict__ emit,   // [S*B, T]
    const int*   __restrict__ label,  // [S, B]
    const float* __restrict__ trans,  // [T, T]  trans[cur, prev]
    float*       __restrict__ out)    // [64 nll | 64*512 path | 64 score] as f32
{
  __shared__ unsigned char s_bp[S_LEN * TAGS];   // 16 KB backpointers (this batch)

  const int b    = blockIdx.x;
  const int lane = threadIdx.x & 31;
  const int wave = threadIdx.x >> 5;

  // Each lane (= cur tag) caches its transition row in VGPRs.
  float tr[TAGS];
#pragma unroll
  for (int p = 0; p < TAGS; ++p) tr[p] = trans[lane * TAGS + p];

  const float te = trans[END_TAG * TAGS + lane];   // trans[end, lane]

  if (wave == 0) {
    // ---------------- alpha (log-sum-exp) recursion ----------------
    float dp = (lane == START_TAG) ? 0.0f : NEG_LOGIT;
    for (int t = 0; t < S_LEN; ++t) {
      const float e = emit[((size_t)t * BATCH + b) * TAGS + lane];
      float vals[TAGS];
#pragma unroll
      for (int p = 0; p < TAGS; ++p) vals[p] = tr[p] + __shfl(dp, p);
      // 4-way interleaved max for ILP
      float m0 = -INFINITY, m1 = -INFINITY, m2 = -INFINITY, m3 = -INFINITY;
#pragma unroll
      for (int p = 0; p < TAGS; p += 4) {
        m0 = fmaxf(m0, vals[p]);     m1 = fmaxf(m1, vals[p + 1]);
        m2 = fmaxf(m2, vals[p + 2]); m3 = fmaxf(m3, vals[p + 3]);
      }
      const float m = fmaxf(fmaxf(m0, m1), fmaxf(m2, m3));
      float s0 = 0.f, s1 = 0.f, s2 = 0.f, s3 = 0.f;
#pragma unroll
      for (int p = 0; p < TAGS; p += 4) {
        s0 += __expf(vals[p] - m);     s1 += __expf(vals[p + 1] - m);
        s2 += __expf(vals[p + 2] - m); s3 += __expf(vals[p + 3] - m);
      }
      dp = m + __logf((s0 + s1) + (s2 + s3)) + e;
    }
    // log partition = logsumexp_prev(dp[prev] + trans[end, prev])
    const float fin = dp + te;
    float fvals[TAGS];
#pragma unroll
    for (int p = 0; p < TAGS; ++p) fvals[p] = __shfl(fin, p);
    float fm = -INFINITY;
#pragma unroll
    for (int p = 0; p < TAGS; ++p) fm = fmaxf(fm, fvals[p]);
    float fs = 0.f;
#pragma unroll
    for (int p = 0; p < TAGS; ++p) fs += __expf(fvals[p] - fm);
    const float logZ = fm + __logf(fs);

    // ---------------- gold path score (lane-strided over S) ----------------
    float acc = 0.0f;
    for (int s = lane; s < S_LEN; s += 32) {
      const int cur  = label[(size_t)s * BATCH + b];
      const int prev = (s == 0) ? START_TAG : label[(size_t)(s - 1) * BATCH + b];
      acc += trans[cur * TAGS + prev] + emit[((size_t)s * BATCH + b) * TAGS + cur];
    }
    if (lane == 0)
      acc += trans[END_TAG * TAGS + label[(size_t)(S_LEN - 1) * BATCH + b]];
#pragma unroll
    for (int off = 16; off > 0; off >>= 1) acc += __shfl_down(acc, off);
    if (lane == 0) out[b] = logZ - acc;               // nll_loss[b]
  } else {
    // ---------------- Viterbi recursion ----------------
    float vdp = (lane == START_TAG) ? 0.0f : NEG_LOGIT;
    for (int t = 0; t < S_LEN; ++t) {
      const float e = emit[((size_t)t * BATCH + b) * TAGS + lane];
      float vals[TAGS];
#pragma unroll
      for (int p = 0; p < TAGS; ++p) vals[p] = tr[p] + __shfl(vdp, p);
      // 4-way interleaved argmax, first-index tie-break like jnp.argmax
      float vb0 = -INFINITY, vb1 = -INFINITY, vb2 = -INFINITY, vb3 = -INFINITY;
      int   ba0 = 0, ba1 = 1, ba2 = 2, ba3 = 3;
#pragma unroll
      for (int p = 0; p < TAGS; p += 4) {
        if (vals[p]     > vb0) { vb0 = vals[p];     ba0 = p;     }
        if (vals[p + 1] > vb1) { vb1 = vals[p + 1]; ba1 = p + 1; }
        if (vals[p + 2] > vb2) { vb2 = vals[p + 2]; ba2 = p + 2; }
        if (vals[p + 3] > vb3) { vb3 = vals[p + 3]; ba3 = p + 3; }
      }
      float vb = vb0; int ba = ba0;
      if (vb1 > vb || (vb1 == vb && ba1 < ba)) { vb = vb1; ba = ba1; }
      if (vb2 > vb || (vb2 == vb && ba2 < ba)) { vb = vb2; ba = ba2; }
      if (vb3 > vb || (vb3 == vb && ba3 < ba)) { vb = vb3; ba = ba3; }
      vdp = vb + e;
      s_bp[t * TAGS + lane] = (unsigned char)ba;      // LDS backpointer
    }
    // final best tag over (vdp + trans[end, prev])
    const float fin = vdp + te;
    float best = -INFINITY; int bestTag = 0;
#pragma unroll
    for (int p = 0; p < TAGS; ++p) {
      const float v = __shfl(fin, p);
      if (v > best) { best = v; bestTag = p; }        // strict > keeps lowest index
    }
    if (lane == 0) {
      out[64 + (size_t)BATCH * S_LEN + b] = best;     // batch_best_score[b]
      // backtrace: path[S-1] = bestTag; path[s] = bp[s+1][path[s+1]]
      float* path = out + 64 + (size_t)b * S_LEN;
      int tag = bestTag;
      for (int s = S_LEN - 1; s >= 0; --s) {
        path[s] = (float)tag;
        tag = (int)s_bp[s * TAGS + tag];
      }
    }
  }
}

// ---------------------------------------------------------------------------
extern "C" void kernel_launch(void* const* d_in, const int* in_sizes, int n_in,
                              void* d_out, int out_size, void* d_ws, size_t ws_size,
                              hipStream_t stream) {
  const float* emb   = (const float*)d_in[0];   // [512, 64, 768] f32
  const int*   lbl   = (const int*)  d_in[1];   // [512, 64] i32
  const float* W     = (const float*)d_in[2];   // [768, 32] f32
  const float* bias  = (const float*)d_in[3];   // [32] f32
  const float* trans = (const float*)d_in[4];   // [32, 32] f32
  float* out  = (float*)d_out;
  float* emit = (float*)d_ws;                   // 32768*32*4 = 4 MB scratch

  // 32768 rows / (8 waves * 16 rows) = 256 blocks of 256 threads
  emit_gemm_kernel<<<256, 256, 0, stream>>>(emb, W, bias, emit);
  // one block (2 waves) per batch element
  crf_kernel<<<BATCH, 64, 0, stream>>>(emit, lbl, trans, out);
}